// SegConvLayer_35192962023976
// MI455X (gfx1250) — compile-verified
//
#include <hip/hip_runtime.h>
#include <hip/hip_bf16.h>
#include <stdint.h>

typedef __attribute__((ext_vector_type(2))) float v2f;
typedef __attribute__((ext_vector_type(8))) float v8f;
typedef __attribute__((ext_vector_type(4))) unsigned int u32x4;
typedef __attribute__((ext_vector_type(4))) int i32x4;
typedef __attribute__((ext_vector_type(8))) int i32x8;

#define KNB 16

__device__ __forceinline__ v8f wmma_f32_16x16x4(v2f a, v2f b, v8f c) {
    // D = A(16x4, f32) * B(4x16, f32) + C(16x16, f32)
    return __builtin_amdgcn_wmma_f32_16x16x4_f32(
        /*neg_a=*/false, a, /*neg_b=*/false, b,
        /*c_mod=*/(short)0, c, /*reuse_a=*/false, /*reuse_b=*/false);
}

// ---------------------------------------------------------------------------
// Tensor Data Mover: 2-D f32 tile (global, row-major) -> LDS, one descriptor.
// Guarded: falls back to cooperative loads if the builtin is unavailable.
// ---------------------------------------------------------------------------
#if __has_builtin(__builtin_amdgcn_tensor_load_to_lds) && \
    __has_builtin(__builtin_amdgcn_s_wait_tensorcnt)
#define HAVE_TDM 1
__device__ __forceinline__ void tdm_load_tile_f32(
    const float* gsrc, uint32_t lds_off,
    uint32_t dim0, uint32_t dim1, uint32_t tile0, uint32_t tile1,
    uint32_t stride0)
{
    uint64_t ga = (uint64_t)(uintptr_t)gsrc;
    u32x4 g0;
    g0[0] = 1u;                                    // count=1 (valid), user mode
    g0[1] = lds_off;                               // lds_addr (bytes)
    g0[2] = (uint32_t)ga;                          // global_addr[31:0]
    g0[3] = (uint32_t)((ga >> 32) & 0x01FFFFFFu)   // global_addr[56:32]
          | (2u << 30);                            // type = 2 ("image")
    uint32_t w[8];
    w[0] = (2u << 16);                             // data_size = 4 B; mask=0
    w[1] = (dim0 & 0xFFFFu) << 16;                 // tensor_dim0 @ bits 79:48
    w[2] = (dim0 >> 16) | ((dim1 & 0xFFFFu) << 16);// ... | tensor_dim1 @ 111:80
    w[3] = (dim1 >> 16) | (tile0 << 16);           // ... | tile_dim0 @ 127:112
    w[4] = tile1 & 0xFFFFu;                        // tile_dim1; tile_dim2 = 0
    w[5] = stride0;                                // tensor_dim0_stride[31:0]
    w[6] = 0;                                      // stride0 hi / dim1_stride lo
    w[7] = 0;                                      // dim1_stride hi
    i32x8 g1;
#pragma unroll
    for (int i = 0; i < 8; ++i) g1[i] = (int)w[i];
    i32x4 z4 = {0, 0, 0, 0};
#if __clang_major__ >= 23
    i32x8 z8 = {0, 0, 0, 0, 0, 0, 0, 0};
    __builtin_amdgcn_tensor_load_to_lds(g0, g1, z4, z4, z8, 0);
#else
    __builtin_amdgcn_tensor_load_to_lds(g0, g1, z4, z4, 0);
#endif
}
#endif

// ---------------------------------------------------------------------------
// Pass 1: geometry features -> win[n][14][16]
// ---------------------------------------------------------------------------
__global__ __launch_bounds__(256) void geo_kernel(
    const float* __restrict__ xyz, const float* __restrict__ cov,
    const float* __restrict__ feats, const int* __restrict__ knn,
    float* __restrict__ win, int npts)
{
    int tid = blockIdx.x * 256 + threadIdx.x;
    if (tid >= npts * KNB) return;
    int n = tid >> 4, k = tid & 15;
    int idx  = knn[n * KNB + k];
    int idx0 = knn[n * KNB];

    float lx = xyz[idx * 3 + 0] - xyz[idx0 * 3 + 0];
    float ly = xyz[idx * 3 + 1] - xyz[idx0 * 3 + 1];
    float lz = xyz[idx * 3 + 2] - xyz[idx0 * 3 + 2];

    const float* fa = feats + (size_t)idx  * 64;
    const float* f0 = feats + (size_t)idx0 * 64;
    float nax = fa[3], nay = fa[4], naz = fa[5];   // n_alpha
    float nmx = f0[3], nmy = f0[4], nmz = f0[5];   // n_miu

    float t9  = sqrtf(lx * lx + ly * ly + lz * lz);
    float inv = 1.0f / fmaxf(t9, 1e-12f);
    float rx = lx * inv, ry = ly * inv, rz = lz * inv;      // r_hat

    float t2 = nmx * rx + nmy * ry + nmz * rz;              // dot(n_miu, r_hat)
    float vx = nmx - t2 * rx, vy = nmy - t2 * ry, vz = nmz - t2 * rz;
    float vi = 1.0f / fmaxf(sqrtf(vx * vx + vy * vy + vz * vz), 1e-12f);
    vx *= vi; vy *= vi; vz *= vi;                           // v_miu

    float wx = ry * vz - rz * vy;
    float wy = rz * vx - rx * vz;
    float wz = rx * vy - ry * vx;                           // cross(r_hat, v_miu)
    float wi = 1.0f / fmaxf(sqrtf(wx * wx + wy * wy + wz * wz), 1e-12f);
    wx *= wi; wy *= wi; wz *= wi;                           // w_miu

    float t1 = nax * nmx + nay * nmy + naz * nmz;
    float t3 = rx * nax + ry * nay + rz * naz;
    float t4 = lx * nmx + ly * nmy + lz * nmz;
    float t6 = nax * vx + nay * vy + naz * vz;
    float t7 = nax * wx + nay * wy + naz * wz;
    float cx = nay * nmz - naz * nmy;
    float cy = naz * nmx - nax * nmz;
    float cz = nax * nmy - nay * nmx;                       // cross(n_alpha, n_miu)
    float t8 = lx * cx + ly * cy + lz * cz;

    const float* Ck = cov + (size_t)idx  * 9;
    const float* C0 = cov + (size_t)idx0 * 9;
    float s1 = lx * (Ck[0] * lx + Ck[1] * ly + Ck[2] * lz)
             + ly * (Ck[3] * lx + Ck[4] * ly + Ck[5] * lz)
             + lz * (Ck[6] * lx + Ck[7] * ly + Ck[8] * lz);
    float s2 = lx * (C0[0] * lx + C0[1] * ly + C0[2] * lz)
             + ly * (C0[3] * lx + C0[4] * ly + C0[5] * lz)
             + lz * (C0[6] * lx + C0[7] * ly + C0[8] * lz);

    float* wp = win + (size_t)n * (14 * KNB);
    wp[ 0 * KNB + k] = lx;  wp[ 1 * KNB + k] = ly;  wp[ 2 * KNB + k] = lz;
    wp[ 3 * KNB + k] = t1;  wp[ 4 * KNB + k] = t2;  wp[ 5 * KNB + k] = t3;
    wp[ 6 * KNB + k] = t4;  wp[ 7 * KNB + k] = t3;  /* t5 == t3 */
    wp[ 8 * KNB + k] = t6;  wp[ 9 * KNB + k] = t7;  wp[10 * KNB + k] = t8;
    wp[11 * KNB + k] = t9;  wp[12 * KNB + k] = s1;  wp[13 * KNB + k] = s2;
}

// ---------------------------------------------------------------------------
// Tiny w-MLP convs (VALU; 8/16-channel matrices too small for WMMA tiles)
// ---------------------------------------------------------------------------
template <int CIN, int COUT>
__global__ __launch_bounds__(256) void conv_w_kernel(
    const float* __restrict__ x, const float* __restrict__ W,
    const float* __restrict__ bias,
    const float* __restrict__ inScale, const float* __restrict__ inShift,
    float* __restrict__ y, float* __restrict__ gSum, float* __restrict__ gSqs,
    int npts)
{
    __shared__ float sS[COUT], sQ[COUT];
    if (threadIdx.x < COUT) { sS[threadIdx.x] = 0.0f; sQ[threadIdx.x] = 0.0f; }
    __syncthreads();

    int tid = blockIdx.x * 256 + threadIdx.x;
    const int per = COUT * KNB;
    int n = tid / per;
    if (n < npts) {
        int r = tid - n * per;
        int o = r >> 4, k = r & 15;
        float acc = bias[o];
        const float* xp = x + (size_t)n * (CIN * KNB);
#pragma unroll
        for (int c = 0; c < CIN; ++c) {
            float xv = xp[c * KNB + k];
            if (inScale) xv = fmaxf(xv * inScale[c] + inShift[c], 0.0f);
            acc += W[o * CIN + c] * xv;
        }
        y[(size_t)n * per + o * KNB + k] = acc;
        atomicAdd(&sS[o], acc);
        atomicAdd(&sQ[o], acc * acc);
    }
    __syncthreads();
    if (threadIdx.x < COUT) {
        atomicAdd(&gSum[threadIdx.x], sS[threadIdx.x]);
        atomicAdd(&gSqs[threadIdx.x], sQ[threadIdx.x]);
    }
}

// ---------------------------------------------------------------------------
// BN stats -> (scale, shift).
// ---------------------------------------------------------------------------
__global__ void bn_finalize_kernel(const float* __restrict__ sum,
                                   const float* __restrict__ sumsq,
                                   const float* __restrict__ g,
                                   const float* __restrict__ b,
                                   float* __restrict__ scale,
                                   float* __restrict__ shift,
                                   int C, float invCount)
{
    int c = threadIdx.x;
    if (c < C) {
        float m = sum[c] * invCount;
        float v = sumsq[c] * invCount - m * m;
        float s = g[c] * rsqrtf(v + 1e-5f);
        scale[c] = s;
        shift[c] = b[c] - m * s;
    }
}

// ---------------------------------------------------------------------------
// Shared WMMA core for the 64x64 feature convs:
//   acc[rt] over 4 row-tiles, B panel cached per k-chunk, 4 independent
//   accumulation chains (hides WMMA RAW latency), A streamed from LDS.
// ---------------------------------------------------------------------------
__device__ __forceinline__ void conv_f_core(
    const float* __restrict__ sW, const float* __restrict__ bw,
    const float* __restrict__ bias, int col, int hi,
    float* __restrict__ yOut, float* __restrict__ sS, float* __restrict__ sQ,
    size_t ybase)
{
    v8f acc[4];
#pragma unroll
    for (int rt = 0; rt < 4; ++rt)
#pragma unroll
        for (int v = 0; v < 8; ++v) acc[rt][v] = bias[rt * 16 + v + hi * 8];

    for (int kc = 0; kc < 16; ++kc) {
        int k0 = kc * 4 + hi * 2;
        v2f b;
        b.x = bw[k0 * KNB + col];
        b.y = bw[(k0 + 1) * KNB + col];
#pragma unroll
        for (int rt = 0; rt < 4; ++rt) {
            v2f a;
            a.x = sW[(rt * 16 + col) * 64 + k0];
            a.y = sW[(rt * 16 + col) * 64 + k0 + 1];
            acc[rt] = wmma_f32_16x16x4(a, b, acc[rt]);
        }
    }
#pragma unroll
    for (int rt = 0; rt < 4; ++rt)
#pragma unroll
        for (int v = 0; v < 8; ++v) {
            int ch = rt * 16 + v + hi * 8;
            float val = acc[rt][v];
            yOut[ybase + ch * KNB + col] = val;
            atomicAdd(&sS[ch], val);
            atomicAdd(&sQ[ch], val * val);
        }
}

// ---------------------------------------------------------------------------
// Feature conv 1 (WMMA): y[n][64][16] = fw1(64x64) * gathered_featsT + fb1
// Weights staged by the Tensor Data Mover (one D#, issued by wave 0).
// ---------------------------------------------------------------------------
__global__ __launch_bounds__(256) void conv_f1_kernel(
    const float* __restrict__ feats, const int* __restrict__ knn,
    const float* __restrict__ W, const float* __restrict__ bias,
    float* __restrict__ y, float* __restrict__ gSum, float* __restrict__ gSqs,
    int npts)
{
    __shared__ float sW[64 * 64];            // 16 KB weights (TDM tile)
    __shared__ float sB[8][64 * KNB];        // 32 KB: per-wave B panel (64x16)
    __shared__ float sS[64], sQ[64];

    int t = threadIdx.x;
    if (t < 64) { sS[t] = 0.0f; sQ[t] = 0.0f; }
#if defined(HAVE_TDM)
    if (t == 0)
        tdm_load_tile_f32(W, (uint32_t)(uintptr_t)&sW[0], 64, 64, 64, 64, 64);
#else
#pragma unroll
    for (int i = 0; i < 16; ++i) sW[i * 256 + t] = W[i * 256 + t];
#endif

    int wave = t >> 5, lane = t & 31;
    int n = blockIdx.x * 8 + wave;
    int k = lane & 15, hi = lane >> 4;

    if (n < npts) {
        int idx = knn[n * KNB + k];
        const float* fp = feats + (size_t)idx * 64 + hi * 32;
        float* bp = &sB[wave][0];
#pragma unroll
        for (int c = 0; c < 32; ++c) bp[(hi * 32 + c) * KNB + k] = fp[c];
    }
#if defined(HAVE_TDM)
    __builtin_amdgcn_s_wait_tensorcnt(0);    // no-op for waves with TENSORcnt==0
#endif
    __syncthreads();

    if (n < npts)
        conv_f_core(sW, &sB[wave][0], bias, lane & 15, hi,
                    y, sS, sQ, (size_t)n * 1024);
    __syncthreads();
    if (t < 64) { atomicAdd(&gSum[t], sS[t]); atomicAdd(&gSqs[t], sQ[t]); }
}

// ---------------------------------------------------------------------------
// Feature conv 2 (WMMA): input = bn_relu(yF1), weights fw2 (TDM-staged).
// ---------------------------------------------------------------------------
__global__ __launch_bounds__(256) void conv_f2_kernel(
    const float* __restrict__ yIn,
    const float* __restrict__ inScale, const float* __restrict__ inShift,
    const float* __restrict__ W, const float* __restrict__ bias,
    float* __restrict__ y, float* __restrict__ gSum, float* __restrict__ gSqs,
    int npts)
{
    __shared__ float sW[64 * 64];
    __shared__ float sB[8][64 * KNB];
    __shared__ float sS[64], sQ[64];

    int t = threadIdx.x;
    if (t < 64) { sS[t] = 0.0f; sQ[t] = 0.0f; }
#if defined(HAVE_TDM)
    if (t == 0)
        tdm_load_tile_f32(W, (uint32_t)(uintptr_t)&sW[0], 64, 64, 64, 64, 64);
#else
#pragma unroll
    for (int i = 0; i < 16; ++i) sW[i * 256 + t] = W[i * 256 + t];
#endif

    int wave = t >> 5, lane = t & 31;
    int n = blockIdx.x * 8 + wave;
    int k = lane & 15, hi = lane >> 4;

    if (n < npts) {
        const float* ip = yIn + (size_t)n * 1024;
        float* bp = &sB[wave][0];
#pragma unroll
        for (int c = 0; c < 32; ++c) {
            int cc = hi * 32 + c;
            bp[cc * KNB + k] =
                fmaxf(ip[cc * KNB + k] * inScale[cc] + inShift[cc], 0.0f);
        }
    }
#if defined(HAVE_TDM)
    __builtin_amdgcn_s_wait_tensorcnt(0);
#endif
    __syncthreads();

    if (n < npts)
        conv_f_core(sW, &sB[wave][0], bias, lane & 15, hi,
                    y, sS, sQ, (size_t)n * 1024);
    __syncthreads();
    if (t < 64) { atomicAdd(&gSum[t], sS[t]); atomicAdd(&gSqs[t], sQ[t]); }
}

// ---------------------------------------------------------------------------
// Combine (WMMA): newB[n][h*16+m] = sum_k bn_relu(f2)[h][k] * bn_relu(w3)[m][k]
// ---------------------------------------------------------------------------
__global__ __launch_bounds__(256) void combine_kernel(
    const float* __restrict__ yF2,
    const float* __restrict__ scF, const float* __restrict__ shF,
    const float* __restrict__ yW3,
    const float* __restrict__ scW, const float* __restrict__ shW,
    float* __restrict__ newB, int npts)
{
    __shared__ float sF[8][64 * KNB];        // 32 KB
    __shared__ float sM[8][16 * KNB];        //  8 KB

    int t = threadIdx.x, wave = t >> 5, lane = t & 31;
    int n = blockIdx.x * 8 + wave;

    if (n < npts) {
        const float* fp = yF2 + (size_t)n * 1024;
        float* fb = &sF[wave][0];
        for (int i = lane; i < 1024; i += 32) {
            int c = i >> 4;
            fb[i] = fmaxf(fp[i] * scF[c] + shF[c], 0.0f);
        }
        const float* wp = yW3 + (size_t)n * 256;
        float* mb = &sM[wave][0];
        for (int i = lane; i < 256; i += 32) {
            int m = i >> 4;
            mb[i] = fmaxf(wp[i] * scW[m] + shW[m], 0.0f);
        }
    }
    __syncthreads();

    if (n < npts) {
        int col = lane & 15, hi = lane >> 4;
        const float* fb = &sF[wave][0];
        const float* mb = &sM[wave][0];
        v8f acc[4];
#pragma unroll
        for (int rt = 0; rt < 4; ++rt)
#pragma unroll
            for (int v = 0; v < 8; ++v) acc[rt][v] = 0.0f;
#pragma unroll
        for (int kc = 0; kc < 4; ++kc) {
            int k0 = kc * 4 + hi * 2;
            v2f b;
            b.x = mb[col * KNB + k0];        // B[kk][m] = w[m][kk]
            b.y = mb[col * KNB + k0 + 1];
#pragma unroll
            for (int rt = 0; rt < 4; ++rt) {
                v2f a;
                a.x = fb[(rt * 16 + col) * KNB + k0];
                a.y = fb[(rt * 16 + col) * KNB + k0 + 1];
                acc[rt] = wmma_f32_16x16x4(a, b, acc[rt]);
            }
        }
#pragma unroll
        for (int rt = 0; rt < 4; ++rt)
#pragma unroll
            for (int v = 0; v < 8; ++v) {
                int h = rt * 16 + v + hi * 8;
                newB[(size_t)n * 1024 + h * 16 + col] = acc[rt][v];
            }
    }
}

// ---------------------------------------------------------------------------
// Final linear (WMMA GEMM): outRaw(N x 64) = newB(N x 1024) * lin_w^T + lin_b
// Block = 8 waves -> 2 point-tiles x 4 col-tiles = 32 points x 64 outputs.
// ---------------------------------------------------------------------------
__global__ __launch_bounds__(256) void final_gemm_kernel(
    const float* __restrict__ newB, const float* __restrict__ linW,
    const float* __restrict__ linB, float* __restrict__ outRaw,
    float* __restrict__ gSum, float* __restrict__ gSqs, int npts)
{
    __shared__ float sS[64], sQ[64];
    int t = threadIdx.x;
    if (t < 64) { sS[t] = 0.0f; sQ[t] = 0.0f; }
    __syncthreads();

    int wave = t >> 5, lane = t & 31;
    int pt = wave >> 2, ct = wave & 3;
    int n0 = blockIdx.x * 32 + pt * 16;
    int col = lane & 15, hi = lane >> 4;

    if (n0 < npts) {
        v8f acc;
#pragma unroll
        for (int v = 0; v < 8; ++v) acc[v] = linB[ct * 16 + col];
        const float* ap = newB + (size_t)(n0 + col) * 1024;        // A row = point
        const float* bp = linW + (size_t)(ct * 16 + col) * 1024;   // B col = out ch
        // Each lane streams a private 4 KB row: prefetch ahead into GL2/WGP$.
        __builtin_prefetch(ap + 256, 0, 1);
        __builtin_prefetch(ap + 768, 0, 1);
        __builtin_prefetch(bp + 256, 0, 1);
        __builtin_prefetch(bp + 768, 0, 1);
        for (int kc = 0; kc < 256; ++kc) {
            int k0 = kc * 4 + hi * 2;
            v2f a, b;
            a.x = ap[k0];     a.y = ap[k0 + 1];
            b.x = bp[k0];     b.y = bp[k0 + 1];
            acc = wmma_f32_16x16x4(a, b, acc);
        }
        float ls = 0.0f, lq = 0.0f;
#pragma unroll
        for (int v = 0; v < 8; ++v) {
            int nn = n0 + v + hi * 8;
            float val = acc[v];
            outRaw[(size_t)nn * 64 + ct * 16 + col] = val;
            ls += val; lq += val * val;
        }
        atomicAdd(&sS[ct * 16 + col], ls);
        atomicAdd(&sQ[ct * 16 + col], lq);
    }
    __syncthreads();
    if (t < 64) { atomicAdd(&gSum[t], sS[t]); atomicAdd(&gSqs[t], sQ[t]); }
}

// ---------------------------------------------------------------------------
// Final BN + ReLU -> d_out
// ---------------------------------------------------------------------------
__global__ __launch_bounds__(256) void final_bn_kernel(
    const float* __restrict__ raw, const float* __restrict__ sc,
    const float* __restrict__ sh, float* __restrict__ out, int total)
{
    int i = blockIdx.x * 256 + threadIdx.x;
    if (i < total) {
        int c = i & 63;
        out[i] = fmaxf(raw[i] * sc[c] + sh[c], 0.0f);
    }
}

// ---------------------------------------------------------------------------
extern "C" void kernel_launch(void* const* d_in, const int* in_sizes, int n_in,
                              void* d_out, int out_size, void* d_ws, size_t ws_size,
                              hipStream_t stream)
{
    const float* xyz   = (const float*)d_in[0];
    const float* cov   = (const float*)d_in[1];
    const float* feats = (const float*)d_in[2];
    const int*   knn   = (const int*)  d_in[3];
    const float* fw1 = (const float*)d_in[4],  *fb1 = (const float*)d_in[5];
    const float* fg1 = (const float*)d_in[6],  *fbe1= (const float*)d_in[7];
    const float* fw2 = (const float*)d_in[8],  *fb2 = (const float*)d_in[9];
    const float* fg2 = (const float*)d_in[10], *fbe2= (const float*)d_in[11];
    const float* ww1 = (const float*)d_in[12], *wb1 = (const float*)d_in[13];
    const float* wg1 = (const float*)d_in[14], *wbe1= (const float*)d_in[15];
    const float* ww2 = (const float*)d_in[16], *wb2 = (const float*)d_in[17];
    const float* wg2 = (const float*)d_in[18], *wbe2= (const float*)d_in[19];
    const float* ww3 = (const float*)d_in[20], *wb3 = (const float*)d_in[21];
    const float* wg3 = (const float*)d_in[22], *wbe3= (const float*)d_in[23];
    const float* linW = (const float*)d_in[24], *linB = (const float*)d_in[25];
    const float* bng = (const float*)d_in[26], *bnb = (const float*)d_in[27];

    const int npts = in_sizes[0] / 3;           // N (=100000)
    const float invNK = 1.0f / ((float)npts * (float)KNB);
    const float invN  = 1.0f / (float)npts;

    float* ws = (float*)d_ws;
    size_t off = 0;
    float* win = ws + off; off += (size_t)npts * 14 * KNB;   // 14x16 / point
    float* y1w = ws + off; off += (size_t)npts * 8  * KNB;   // 8x16
    float* y2w = ws + off; off += (size_t)npts * 8  * KNB;   // 8x16
    float* y3w = ws + off; off += (size_t)npts * 16 * KNB;   // 16x16
    float* yF1 = ws + off; off += (size_t)npts * 64 * KNB;   // 64x16
    float* yF2 = ws + off; off += (size_t)npts * 64 * KNB;   // 64x16
    float* stats = ws + off; off += 6 * 256;
    float* newB   = yF1;   // yF1 dead after conv_f2 consumed it
    float* outRaw = win;   // win dead after conv_w1

#define SUM(L) (stats + (L) * 256)
#define SQS(L) (stats + (L) * 256 + 64)
#define SCL(L) (stats + (L) * 256 + 128)
#define SHF(L) (stats + (L) * 256 + 192)

    hipMemsetAsync(stats, 0, 6 * 256 * sizeof(float), stream);

    // 1. geometry features
    geo_kernel<<<(npts * KNB + 255) / 256, 256, 0, stream>>>(
        xyz, cov, feats, knn, win, npts);

    // 2-4. w-MLP: 14 -> 8 -> 8 -> 16, BN between
    conv_w_kernel<14, 8><<<(npts * 8 * KNB + 255) / 256, 256, 0, stream>>>(
        win, ww1, wb1, nullptr, nullptr, y1w, SUM(0), SQS(0), npts);
    bn_finalize_kernel<<<1, 64, 0, stream>>>(SUM(0), SQS(0), wg1, wbe1,
                                             SCL(0), SHF(0), 8, invNK);
    conv_w_kernel<8, 8><<<(npts * 8 * KNB + 255) / 256, 256, 0, stream>>>(
        y1w, ww2, wb2, SCL(0), SHF(0), y2w, SUM(1), SQS(1), npts);
    bn_finalize_kernel<<<1, 64, 0, stream>>>(SUM(1), SQS(1), wg2, wbe2,
                                             SCL(1), SHF(1), 8, invNK);
    conv_w_kernel<8, 16><<<(npts * 16 * KNB + 255) / 256, 256, 0, stream>>>(
        y2w, ww3, wb3, SCL(1), SHF(1), y3w, SUM(2), SQS(2), npts);
    bn_finalize_kernel<<<1, 64, 0, stream>>>(SUM(2), SQS(2), wg3, wbe3,
                                             SCL(2), SHF(2), 16, invNK);

    // 5-6. feature MLP: 64 -> 64 -> 64 (WMMA, TDM-staged weights)
    conv_f1_kernel<<<(npts + 7) / 8, 256, 0, stream>>>(
        feats, knn, fw1, fb1, yF1, SUM(3), SQS(3), npts);
    bn_finalize_kernel<<<1, 64, 0, stream>>>(SUM(3), SQS(3), fg1, fbe1,
                                             SCL(3), SHF(3), 64, invNK);
    conv_f2_kernel<<<(npts + 7) / 8, 256, 0, stream>>>(
        yF1, SCL(3), SHF(3), fw2, fb2, yF2, SUM(4), SQS(4), npts);
    bn_finalize_kernel<<<1, 64, 0, stream>>>(SUM(4), SQS(4), fg2, fbe2,
                                             SCL(4), SHF(4), 64, invNK);

    // 7. combine f x w^T  (WMMA)
    combine_kernel<<<(npts + 7) / 8, 256, 0, stream>>>(
        yF2, SCL(4), SHF(4), y3w, SCL(2), SHF(2), newB, npts);

    // 8. final linear 1024 -> 64 (WMMA GEMM) + stats
    final_gemm_kernel<<<(npts + 31) / 32, 256, 0, stream>>>(
        newB, linW, linB, outRaw, SUM(5), SQS(5), npts);
    bn_finalize_kernel<<<1, 64, 0, stream>>>(SUM(5), SQS(5), bng, bnb,
                                             SCL(5), SHF(5), 64, invN);

    // 9. final BN + ReLU
    final_bn_kernel<<<(npts * 64 + 255) / 256, 256, 0, stream>>>(
        outRaw, SCL(5), SHF(5), (float*)d_out, npts * 64);

#undef SUM
#undef SQS
#undef SCL
#undef SHF
    (void)n_in; (void)out_size; (void)ws_size;
}